// NodeModel_47974784696393
// MI455X (gfx1250) — compile-verified
//
#include <hip/hip_runtime.h>
#include <hip/hip_bf16.h>

#define HID 128
#define WGRP 288                 // dwords per k-group (64 groups); 288%64==32 -> conflict-free b64
#define WMAT (64 * WGRP)         // dwords per staged 128x128 weight matrix (72 KB)
#define SROW 130                 // padded scratch row stride (dwords)
#define STILE (16 * SROW)        // scratch dwords per wave

typedef __attribute__((ext_vector_type(2))) float v2f;
typedef __attribute__((ext_vector_type(8))) float v8f;

__device__ __forceinline__ float softplus_f(float x) {
    // stable softplus with HW transcendentals: max(x,0) + log(1 + exp(-|x|))
    return fmaxf(x, 0.0f) + __logf(1.0f + __expf(-fabsf(x)));
}

// ---------------------------------------------------------------------------
// Zero-fill (float4-vectorized)
// ---------------------------------------------------------------------------
__global__ void zero_kernel(float4* __restrict__ p, int n4) {
    int i = blockIdx.x * blockDim.x + threadIdx.x;
    if (i < n4) p[i] = make_float4(0.f, 0.f, 0.f, 0.f);
}

// ---------------------------------------------------------------------------
// biasc[j] = b_c[j] + sum_k global_feat[k] * W_c[256+k][j]   (global batch == 1)
// ---------------------------------------------------------------------------
__global__ void biasc_kernel(const float* __restrict__ g,
                             const float* __restrict__ Wc,
                             const float* __restrict__ bc,
                             float* __restrict__ biasc) {
    int j = threadIdx.x;
    float s = bc[j];
    for (int k = 0; k < HID; ++k)
        s += g[k] * Wc[(size_t)(2 * HID + k) * HID + j];
    biasc[j] = s;
}

// ---------------------------------------------------------------------------
// Scatter-add edge features into node slots (L2 float atomics)
// one thread per (edge, 4 features)
// ---------------------------------------------------------------------------
__global__ void scatter_kernel(const float* __restrict__ ef,
                               const int* __restrict__ row,
                               float* __restrict__ aggr, int n_quads) {
    int gid = blockIdx.x * blockDim.x + threadIdx.x;
    if (gid >= n_quads) return;
    int e = gid >> 5;            // / (HID/4)
    int j = (gid & 31) * 4;      // feature base
    const float4 v = *(const float4*)(ef + (size_t)gid * 4);
    float* dst = aggr + (size_t)row[e] * HID + j;
    atomicAdd(dst + 0, v.x);
    atomicAdd(dst + 1, v.y);
    atomicAdd(dst + 2, v.z);
    atomicAdd(dst + 3, v.w);
}

// ---------------------------------------------------------------------------
// Stage a 128x128 row-major weight into fragment-swizzled LDS:
//   lds[(k>>1)*WGRP + 2*n + (k&1)] = W[k][n]
// Vectorized: read a float4 from rows 2g and 2g+1, interleave, write two
// contiguous 16B-aligned float4 LDS stores.
// ---------------------------------------------------------------------------
__device__ __forceinline__ void stage_weight(float* dst, const float* __restrict__ src,
                                             int tid, int nthreads) {
    for (int idx = tid; idx < 64 * 32; idx += nthreads) {  // 64 k-pairs x 32 n-quads
        int g = idx >> 5;
        int nq = idx & 31;
        const float4 r0 = *(const float4*)(src + (size_t)(2 * g) * HID + nq * 4);
        const float4 r1 = *(const float4*)(src + (size_t)(2 * g + 1) * HID + nq * 4);
        float4 lo = make_float4(r0.x, r1.x, r0.y, r1.y);
        float4 hi = make_float4(r0.z, r1.z, r0.w, r1.w);
        float* d = dst + g * WGRP + nq * 8;
        *(float4*)(d + 0) = lo;
        *(float4*)(d + 4) = hi;
    }
}

// ---------------------------------------------------------------------------
// WMMA fragment conventions (V_WMMA_F32_16X16X4_F32, full f32 precision)
//   A 16x4: lane l holds row (l&15), float2 at col kb + 2*(l>>4)
//   B 4x16: lane l holds (W[kc][n], W[kc+1][n]) with kc = kb+2*(l>>4), n = (l&15)+16t
//   C 16x16: vgpr i of lane l -> row i + 8*(l>>4), col (l&15)
// ---------------------------------------------------------------------------
__device__ __forceinline__ void init_bias(const float* __restrict__ b, int l15,
                                          v8f acc[8]) {
#pragma unroll
    for (int t = 0; t < 8; ++t) {
        float bv = b[t * 16 + l15];
        v8f a;
#pragma unroll
        for (int i = 0; i < 8; ++i) a[i] = bv;
        acc[t] = a;
    }
}

__device__ __forceinline__ void wmma8(const v2f a, const float* __restrict__ wp,
                                      v8f acc[8]) {
    v2f bf[8];
#pragma unroll
    for (int t = 0; t < 8; ++t) bf[t] = *(const v2f*)(wp + t * 32);
#pragma unroll
    for (int t = 0; t < 8; ++t)
        acc[t] = __builtin_amdgcn_wmma_f32_16x16x4_f32(
            false, a, false, bf[t], (short)0, acc[t], false, false);
}

// arow: this lane's 128-float activation row (global or LDS); ldsW: swizzled weight
__device__ __forceinline__ void gemm_tile(const float* __restrict__ arow,
                                          const float* __restrict__ ldsW,
                                          int half, int l15, v8f acc[8]) {
    const float* wbase = ldsW + half * WGRP + l15 * 2;
    v2f a_cur = *(const v2f*)(arow + half * 2);
#pragma unroll 4
    for (int kb = 0; kb < HID - 4; kb += 4) {
        v2f a = a_cur;
        a_cur = *(const v2f*)(arow + (kb + 4) + half * 2);  // unconditional prefetch
        wmma8(a, wbase + (kb >> 1) * WGRP, acc);
    }
    wmma8(a_cur, wbase + ((HID - 4) >> 1) * WGRP, acc);  // peeled final k-step
}

// ---------------------------------------------------------------------------
// combined = x @ Wc[0:128] + aggr @ Wc[128:256] + biasc
// 256 threads = 8 waves; each wave owns 16 node rows. LDS: 2 x 72 KB weights.
// ---------------------------------------------------------------------------
__global__ void combine_kernel(const float* __restrict__ x,
                               const float* __restrict__ aggr,
                               const float* __restrict__ Wc,
                               const float* __restrict__ biasc,
                               float* __restrict__ combined, int n_nodes) {
    extern __shared__ float lds[];
    float* ldsWx = lds;
    float* ldsWa = lds + WMAT;

    const int tid = threadIdx.x;
    stage_weight(ldsWx, Wc, tid, blockDim.x);
    stage_weight(ldsWa, Wc + HID * HID, tid, blockDim.x);
    __syncthreads();

    const int lane = tid & 31;
    const int wave = tid >> 5;
    const int half = lane >> 4;
    const int l15 = lane & 15;

    const int row_base = blockIdx.x * 128 + wave * 16;
    int ar = row_base + l15;
    if (ar >= n_nodes) ar = n_nodes - 1;  // clamp tail reads; stores masked
    const float* xrow = x + (size_t)ar * HID;
    const float* grow = aggr + (size_t)ar * HID;

    v8f acc[8];
    init_bias(biasc, l15, acc);
    gemm_tile(xrow, ldsWx, half, l15, acc);
    gemm_tile(grow, ldsWa, half, l15, acc);

#pragma unroll
    for (int t = 0; t < 8; ++t) {
#pragma unroll
        for (int i = 0; i < 8; ++i) {
            int r = row_base + half * 8 + i;
            if (r < n_nodes)
                combined[(size_t)r * HID + t * 16 + l15] = acc[t][i];
        }
    }
}

// ---------------------------------------------------------------------------
// out = softplus(softplus(combined@W1+b1)@W2+b2)@W3 + b3
// LDS: 3 x 72 KB swizzled weights + 8 waves x (16*130*4 B) scratch ~= 281 KB.
// Inter-layer C->A re-layout via wave-private padded LDS scratch (per-wave DS
// ordering makes barriers unnecessary).
// ---------------------------------------------------------------------------
__global__ void mlp_kernel(const float* __restrict__ combined,
                           const float* __restrict__ W1, const float* __restrict__ b1,
                           const float* __restrict__ W2, const float* __restrict__ b2,
                           const float* __restrict__ W3, const float* __restrict__ b3,
                           float* __restrict__ out, int n_nodes) {
    extern __shared__ float lds[];
    float* ldsW1 = lds;
    float* ldsW2 = lds + WMAT;
    float* ldsW3 = lds + 2 * WMAT;
    float* scratch = lds + 3 * WMAT;

    const int tid = threadIdx.x;
    stage_weight(ldsW1, W1, tid, blockDim.x);
    stage_weight(ldsW2, W2, tid, blockDim.x);
    stage_weight(ldsW3, W3, tid, blockDim.x);
    __syncthreads();

    const int lane = tid & 31;
    const int wave = tid >> 5;
    const int half = lane >> 4;
    const int l15 = lane & 15;
    float* ws = scratch + wave * STILE;  // wave-private 16 x 128 (stride 130)

    const int row_base = blockIdx.x * 128 + wave * 16;
    int ar = row_base + l15;
    if (ar >= n_nodes) ar = n_nodes - 1;
    const float* crow = combined + (size_t)ar * HID;

    v8f acc[8];

    // layer 1
    init_bias(b1, l15, acc);
    gemm_tile(crow, ldsW1, half, l15, acc);
#pragma unroll
    for (int t = 0; t < 8; ++t)
#pragma unroll
        for (int i = 0; i < 8; ++i)
            ws[(half * 8 + i) * SROW + t * 16 + l15] = softplus_f(acc[t][i]);

    // layer 2 (A fragments from wave-private scratch, row = l15)
    init_bias(b2, l15, acc);
    gemm_tile(ws + l15 * SROW, ldsW2, half, l15, acc);
#pragma unroll
    for (int t = 0; t < 8; ++t)
#pragma unroll
        for (int i = 0; i < 8; ++i)
            ws[(half * 8 + i) * SROW + t * 16 + l15] = softplus_f(acc[t][i]);

    // layer 3 (linear) -> global output
    init_bias(b3, l15, acc);
    gemm_tile(ws + l15 * SROW, ldsW3, half, l15, acc);
#pragma unroll
    for (int t = 0; t < 8; ++t) {
#pragma unroll
        for (int i = 0; i < 8; ++i) {
            int r = row_base + half * 8 + i;
            if (r < n_nodes)
                out[(size_t)r * HID + t * 16 + l15] = acc[t][i];
        }
    }
}

// ---------------------------------------------------------------------------
extern "C" void kernel_launch(void* const* d_in, const int* in_sizes, int n_in,
                              void* d_out, int out_size, void* d_ws, size_t ws_size,
                              hipStream_t stream) {
    const float* x  = (const float*)d_in[0];
    const int*   ei = (const int*)d_in[1];   // edge_index [2, E]; row = first E entries
    const float* ef = (const float*)d_in[2];
    const float* g  = (const float*)d_in[3];
    const float* Wc = (const float*)d_in[4];
    const float* bc = (const float*)d_in[5];
    const float* W1 = (const float*)d_in[6];
    const float* b1 = (const float*)d_in[7];
    const float* W2 = (const float*)d_in[8];
    const float* b2 = (const float*)d_in[9];
    const float* W3 = (const float*)d_in[10];
    const float* b3 = (const float*)d_in[11];

    const int n_nodes = in_sizes[0] / HID;  // 100000
    const int n_edges = in_sizes[2] / HID;  // 600000

    float* aggr     = (float*)d_ws;
    float* combined = aggr + (size_t)n_nodes * HID;
    float* biasc    = combined + (size_t)n_nodes * HID;
    float* out      = (float*)d_out;

    // 1) zero edge aggregation buffer
    {
        int n4 = n_nodes * HID / 4;
        zero_kernel<<<(n4 + 255) / 256, 256, 0, stream>>>((float4*)aggr, n4);
    }
    // 2) fold global_feat @ Wc_g + b_c into a single 128-vector
    biasc_kernel<<<1, HID, 0, stream>>>(g, Wc, bc, biasc);
    // 3) scatter-add edge features (float4 per thread)
    {
        int n_quads = n_edges * (HID / 4);  // 19.2M
        scatter_kernel<<<(n_quads + 255) / 256, 256, 0, stream>>>(ef, ei, aggr,
                                                                  n_quads);
    }
    // 4) combine GEMM (x@Wc_x + aggr@Wc_a + biasc), f32 WMMA
    {
        int blocks = (n_nodes + 127) / 128;
        size_t ldsB = (size_t)2 * WMAT * sizeof(float);  // 144 KB
        combine_kernel<<<blocks, 256, ldsB, stream>>>(x, aggr, Wc, biasc, combined,
                                                      n_nodes);
    }
    // 5) fused 3-layer MLP, f32 WMMA, swizzled weights + scratch in LDS
    {
        int blocks = (n_nodes + 127) / 128;
        size_t ldsB = (size_t)(3 * WMAT + 8 * STILE) * sizeof(float);  // ~281 KB
        mlp_kernel<<<blocks, 256, ldsB, stream>>>(combined, W1, b1, W2, b2, W3, b3,
                                                  out, n_nodes);
    }
}